// AlphaGCN_58978490909200
// MI455X (gfx1250) — compile-verified
//
#include <hip/hip_runtime.h>
#include <hip/hip_bf16.h>

#define N_NODES 50000
#define E_EDGES 600000
#define B_GR    64
#define FNODE   64
#define HDIM    128
#define NH      (N_NODES * HDIM)   // 6,400,000
#define BSTRIDE 132                 // LDS row stride (floats), 8-bank skew/2 rows

typedef float v2f __attribute__((ext_vector_type(2)));
typedef float v8f __attribute__((ext_vector_type(8)));

// ---------------------------------------------------------------------------
// degree / dis
// ---------------------------------------------------------------------------
__global__ void deg_kernel(const int* __restrict__ dst, float* __restrict__ deg) {
    int e = blockIdx.x * blockDim.x + threadIdx.x;
    if (e >= E_EDGES) return;
    atomicAdd(&deg[dst[e]], 1.0f);
}

__global__ void dis_kernel(float* __restrict__ dis) {
    int n = blockIdx.x * blockDim.x + threadIdx.x;
    if (n >= N_NODES) return;
    dis[n] = rsqrtf(dis[n] + 1.0f);
}

// ---------------------------------------------------------------------------
// Row-stripe fp32 WMMA GEMM with LDS-staged weights:
//   C[M,128] = A[M,K] @ B[K,128] (+bias)(+relu)
// The whole B matrix (<=64KB) is cooperatively loaded into LDS once per
// block; each wave then computes a 16x128 stripe (8 accumulators) with
// V_WMMA_F32_16X16X4_F32, feeding B fragments from ds_load and one
// global_load_b64 A fragment per k-step (reused by 8 WMMAs).
// A-frag (16x4 f32): lanes 0-15 -> M=lane, VGPR0=K0, VGPR1=K1;
//                    lanes 16-31 -> M=lane-16, VGPR0=K2, VGPR1=K3.
// B-frag (4x16 f32): mirrored. C/D: VGPR v -> M=v (lanes<16) / M=8+v.
// ---------------------------------------------------------------------------
__global__ void gemm_wmma_f32(const float* __restrict__ A,
                              const float* __restrict__ Bm,
                              const float* __restrict__ bias,
                              float* __restrict__ C,
                              int M, int K, int do_relu) {
    __shared__ float Bs[HDIM * BSTRIDE];     // 128 rows max, padded stride

    // Cooperative stage of B[K,128] into LDS (float4, coalesced).
    for (int i = threadIdx.x; i < K * (HDIM / 4); i += blockDim.x) {
        int r  = i >> 5;                     // row (32 float4 per row)
        int c4 = i & 31;
        float4 v = ((const float4*)(Bm + (size_t)r * HDIM))[c4];
        *(float4*)&Bs[r * BSTRIDE + c4 * 4] = v;
    }
    __syncthreads();

    int wave = blockIdx.x * (blockDim.x >> 5) + (threadIdx.x >> 5);
    int row0 = wave * 16;
    if (row0 >= M) return;                   // wave-uniform: EXEC stays all-1s

    int lane = threadIdx.x & 31;
    int half = lane >> 4;                    // 0 or 1
    int l    = lane & 15;

    const float* Arow = A + (size_t)(row0 + l) * K;

    v8f c[8];
#pragma unroll
    for (int j = 0; j < 8; ++j) c[j] = (v8f){};

    for (int k = 0; k < K; k += 4) {
        int ka = k + 2 * half;
        v2f a;
        a.x = Arow[ka];
        a.y = Arow[ka + 1];
        const float* B0 = &Bs[ka * BSTRIDE + l];
        const float* B1 = B0 + BSTRIDE;
#pragma unroll
        for (int j = 0; j < 8; ++j) {
            v2f b;
            b.x = B0[j * 16];
            b.y = B1[j * 16];
            c[j] = __builtin_amdgcn_wmma_f32_16x16x4_f32(
                       /*neg_a=*/false, a, /*neg_b=*/false, b,
                       /*c_mod=*/(short)0, c[j],
                       /*reuse_a=*/false, /*reuse_b=*/false);
        }
    }

#pragma unroll
    for (int j = 0; j < 8; ++j) {
        int   col = j * 16 + l;
        float bv  = bias ? bias[col] : 0.0f;
#pragma unroll
        for (int v = 0; v < 8; ++v) {
            int   row = row0 + v + 8 * half;
            float val = c[j][v] + bv;
            if (do_relu) val = fmaxf(val, 0.0f);
            C[(size_t)row * HDIM + col] = val;
        }
    }
}

// ---------------------------------------------------------------------------
// msg = dis[src] * t[src]; agg[dst] += msg   (float4 gather, float atomics)
// thread = (edge, 4-feature chunk)
// ---------------------------------------------------------------------------
__global__ void edge_scatter_kernel(const int* __restrict__ src,
                                    const int* __restrict__ dst,
                                    const float* __restrict__ dis,
                                    const float* __restrict__ t,
                                    float* __restrict__ agg) {
    int tid = blockIdx.x * blockDim.x + threadIdx.x;
    const int CH = HDIM / 4;                 // 32 chunks
    if (tid >= E_EDGES * CH) return;
    int e  = tid / CH;
    int f4 = tid - e * CH;
    int s  = src[e];
    int d  = dst[e];
    float sc = dis[s];
    float4 v = *((const float4*)(t + (size_t)s * HDIM) + f4);
    float* a = agg + (size_t)d * HDIM + f4 * 4;
    atomicAdd(a + 0, sc * v.x);
    atomicAdd(a + 1, sc * v.y);
    atomicAdd(a + 2, sc * v.z);
    atomicAdd(a + 3, sc * v.w);
}

// ---------------------------------------------------------------------------
// pre = dis*agg + dis^2*t + conv_b  (in place in agg), fused with BN partial
// statistics. blockDim=256 => grid stride is a multiple of HDIM, so each
// thread touches a single fixed feature f = tid & 127.
// ---------------------------------------------------------------------------
__global__ void combine_stats_kernel(const float* __restrict__ dis,
                                     const float* __restrict__ t,
                                     const float* __restrict__ cb,
                                     float* __restrict__ agg,
                                     float* __restrict__ sums,
                                     float* __restrict__ sqs) {
    int f = threadIdx.x & (HDIM - 1);
    float bias = cb[f];
    float s = 0.0f, q = 0.0f;
    int stride = gridDim.x * blockDim.x;
    for (int idx = blockIdx.x * blockDim.x + threadIdx.x; idx < NH; idx += stride) {
        int   n   = idx >> 7;                // idx / HDIM
        float di  = dis[n];
        float pre = di * agg[idx] + di * di * t[idx] + bias;
        agg[idx]  = pre;
        s += pre;
        q += pre * pre;
    }
    __shared__ float ls[256];
    __shared__ float lq[256];
    ls[threadIdx.x] = s;
    lq[threadIdx.x] = q;
    __syncthreads();
    if (threadIdx.x < 128) {
        atomicAdd(&sums[f], ls[threadIdx.x] + ls[threadIdx.x + 128]);
        atomicAdd(&sqs[f],  lq[threadIdx.x] + lq[threadIdx.x + 128]);
    }
}

__global__ void bn_finalize_kernel(const float* __restrict__ sums,
                                   const float* __restrict__ sqs,
                                   float* __restrict__ mean,
                                   float* __restrict__ rstd) {
    int f = threadIdx.x;
    if (f >= HDIM) return;
    float m = sums[f] / (float)N_NODES;
    mean[f] = m;
    rstd[f] = rsqrtf(sqs[f] / (float)N_NODES - m * m + 1e-5f);
}

// h = relu(gamma*(pre-mean)*rstd + beta) (+ old h if add_res)
__global__ void bn_apply_kernel(const float* __restrict__ pre,
                                const float* __restrict__ mean,
                                const float* __restrict__ rstd,
                                const float* __restrict__ gamma,
                                const float* __restrict__ beta,
                                float* __restrict__ h, int add_res) {
    int idx = blockIdx.x * blockDim.x + threadIdx.x;
    if (idx >= NH) return;
    int f = idx & (HDIM - 1);
    float v = gamma[f] * (pre[idx] - mean[f]) * rstd[f] + beta[f];
    v = fmaxf(v, 0.0f);
    if (add_res) v += h[idx];
    h[idx] = v;
}

// ---------------------------------------------------------------------------
// mean pool over graphs
// ---------------------------------------------------------------------------
__global__ void count_kernel(const int* __restrict__ batch_idx,
                             float* __restrict__ counts) {
    int n = blockIdx.x * blockDim.x + threadIdx.x;
    if (n >= N_NODES) return;
    atomicAdd(&counts[batch_idx[n]], 1.0f);
}

__global__ void pool_sum_kernel(const int* __restrict__ batch_idx,
                                const float* __restrict__ h,
                                float* __restrict__ gemb) {
    int tid = blockIdx.x * blockDim.x + threadIdx.x;
    const int CH = HDIM / 4;
    if (tid >= N_NODES * CH) return;
    int n  = tid / CH;
    int f4 = tid - n * CH;
    int b  = batch_idx[n];
    float4 v = *((const float4*)(h + (size_t)n * HDIM) + f4);
    float* g = gemb + (size_t)b * HDIM + f4 * 4;
    atomicAdd(g + 0, v.x);
    atomicAdd(g + 1, v.y);
    atomicAdd(g + 2, v.z);
    atomicAdd(g + 3, v.w);
}

__global__ void pool_div_kernel(float* __restrict__ gemb,
                                const float* __restrict__ counts) {
    int idx = blockIdx.x * blockDim.x + threadIdx.x;
    if (idx >= B_GR * HDIM) return;
    gemb[idx] /= fmaxf(counts[idx / HDIM], 1.0f);
}

// ---------------------------------------------------------------------------
// head MLPs, one thread per graph (B=64)
// ---------------------------------------------------------------------------
__global__ void head_kernel(const float* __restrict__ gemb,
                            const float* __restrict__ gfeat,
                            const float* __restrict__ gfW1, const float* __restrict__ gfb1,
                            const float* __restrict__ gfW2, const float* __restrict__ gfb2,
                            const float* __restrict__ pW1,  const float* __restrict__ pb1,
                            const float* __restrict__ pW2,  const float* __restrict__ pb2,
                            const float* __restrict__ pW3,  const float* __restrict__ pb3,
                            float* __restrict__ out) {
    int b = threadIdx.x;
    if (b >= B_GR) return;

    float gfh[64];
    for (int j = 0; j < 64; ++j) {
        float s = gfb1[j];
        for (int k = 0; k < 7; ++k) s += gfeat[b * 7 + k] * gfW1[k * 64 + j];
        gfh[j] = fmaxf(s, 0.0f);
    }
    float gfo[32];
    for (int j = 0; j < 32; ++j) {
        float s = gfb2[j];
        for (int k = 0; k < 64; ++k) s += gfh[k] * gfW2[k * 32 + j];
        gfo[j] = s;
    }
    float z1[64];
    for (int j = 0; j < 64; ++j) {
        float s = pb1[j];
        for (int k = 0; k < 128; ++k) s += gemb[b * 128 + k] * pW1[k * 64 + j];
        for (int k = 0; k < 32;  ++k) s += gfo[k] * pW1[(128 + k) * 64 + j];
        z1[j] = fmaxf(s, 0.0f);
    }
    float z2[32];
    for (int j = 0; j < 32; ++j) {
        float s = pb2[j];
        for (int k = 0; k < 64; ++k) s += z1[k] * pW2[k * 32 + j];
        z2[j] = fmaxf(s, 0.0f);
    }
    float s = pb3[0];
    for (int k = 0; k < 32; ++k) s += z2[k] * pW3[k];
    out[b] = s;
}

// ---------------------------------------------------------------------------
extern "C" void kernel_launch(void* const* d_in, const int* in_sizes, int n_in,
                              void* d_out, int out_size, void* d_ws, size_t ws_size,
                              hipStream_t stream) {
    const float* x        = (const float*)d_in[0];
    const int*   eidx     = (const int*)  d_in[1];
    const int*   batchidx = (const int*)  d_in[2];
    const float* gfeat    = (const float*)d_in[3];
    const float* W_emb    = (const float*)d_in[4];
    const float* b_emb    = (const float*)d_in[5];
    const float* conv_W   = (const float*)d_in[6];
    const float* conv_b   = (const float*)d_in[7];
    const float* bn_gamma = (const float*)d_in[8];
    const float* bn_beta  = (const float*)d_in[9];
    const float* gfW1     = (const float*)d_in[10];
    const float* gfb1     = (const float*)d_in[11];
    const float* gfW2     = (const float*)d_in[12];
    const float* gfb2     = (const float*)d_in[13];
    const float* pW1      = (const float*)d_in[14];
    const float* pb1      = (const float*)d_in[15];
    const float* pW2      = (const float*)d_in[16];
    const float* pb2      = (const float*)d_in[17];
    const float* pW3      = (const float*)d_in[18];
    const float* pb3      = (const float*)d_in[19];
    float* out = (float*)d_out;

    // Workspace layout (floats)
    float* ws     = (float*)d_ws;
    float* dis    = ws;                 // N (also used as deg accumulator)
    float* h      = ws + 50048;         // N*H
    float* t      = h  + NH;            // N*H
    float* agg    = t  + NH;            // N*H (becomes "pre" after combine)
    float* mean   = agg + NH;           // H
    float* rstd   = mean + HDIM;        // H
    float* bnsum  = rstd + HDIM;        // H
    float* bnsq   = bnsum + HDIM;       // H
    float* gemb   = bnsq + HDIM;        // B*H
    float* counts = gemb + B_GR * HDIM; // B

    const int* src = eidx;
    const int* dst = eidx + E_EDGES;

    // degree -> dis
    hipMemsetAsync(dis, 0, N_NODES * sizeof(float), stream);
    deg_kernel<<<(E_EDGES + 255) / 256, 256, 0, stream>>>(dst, dis);
    dis_kernel<<<(N_NODES + 255) / 256, 256, 0, stream>>>(dis);

    const int row_tiles   = N_NODES / 16;          // 3125
    const int gemm_blocks = (row_tiles + 3) / 4;   // 4 waves/block

    // h = relu(x @ W_emb + b_emb)    [50000,64]@[64,128]
    gemm_wmma_f32<<<gemm_blocks, 128, 0, stream>>>(x, W_emb, b_emb, h,
                                                   N_NODES, FNODE, 1);

    for (int i = 0; i < 3; ++i) {
        // t = h @ conv_W[i]          [50000,128]@[128,128]
        gemm_wmma_f32<<<gemm_blocks, 128, 0, stream>>>(
            h, conv_W + (size_t)i * HDIM * HDIM, nullptr, t, N_NODES, HDIM, 0);

        // agg = scatter_add(dis[src] * t[src])
        hipMemsetAsync(agg, 0, (size_t)NH * sizeof(float), stream);
        edge_scatter_kernel<<<(E_EDGES * (HDIM / 4) + 255) / 256, 256, 0, stream>>>(
            src, dst, dis, t, agg);

        // pre = dis*agg + dis^2*t + conv_b[i]  (in place) + BN partial stats
        hipMemsetAsync(bnsum, 0, 2 * HDIM * sizeof(float), stream);
        combine_stats_kernel<<<2048, 256, 0, stream>>>(
            dis, t, conv_b + i * HDIM, agg, bnsum, bnsq);
        bn_finalize_kernel<<<1, 128, 0, stream>>>(bnsum, bnsq, mean, rstd);

        // h = relu(bn(pre)) (+ residual for i>0)
        bn_apply_kernel<<<(NH + 255) / 256, 256, 0, stream>>>(
            agg, mean, rstd, bn_gamma + i * HDIM, bn_beta + i * HDIM, h, i > 0);
    }

    // global mean pool
    hipMemsetAsync(gemb, 0, (B_GR * HDIM + B_GR) * sizeof(float), stream);
    count_kernel<<<(N_NODES + 255) / 256, 256, 0, stream>>>(batchidx, counts);
    pool_sum_kernel<<<(N_NODES * (HDIM / 4) + 255) / 256, 256, 0, stream>>>(
        batchidx, h, gemb);
    pool_div_kernel<<<(B_GR * HDIM + 255) / 256, 256, 0, stream>>>(gemb, counts);

    // head MLPs -> out[B]
    head_kernel<<<1, 64, 0, stream>>>(gemb, gfeat, gfW1, gfb1, gfW2, gfb2,
                                      pW1, pb1, pW2, pb2, pW3, pb3, out);
}